// torchSOM_7164005449814
// MI455X (gfx1250) — compile-verified
//
#include <hip/hip_runtime.h>
#include <stdint.h>

// FlowSOM online training, MI455X (gfx1250, wave32).
// One persistent 1024-thread workgroup; thread m owns node m (32 f32 in VGPRs).
// x-rows are prefetched 3 iterations ahead with the CDNA5 async global->LDS
// engine (ASYNCcnt / s_wait_asynccnt); argmin via wave32 shuffle butterfly.

#define D        32
#define MNODES   1024
#define NITER    2000
#define PF       3      // prefetch depth
#define NBUF     4      // LDS ring buffers (must be > PF, power of 2)

__device__ __forceinline__ unsigned long long shfl_xor_u64(unsigned long long v, int m) {
  int lo = __shfl_xor((int)(unsigned)(v & 0xffffffffull), m, 32);
  int hi = __shfl_xor((int)(unsigned)(v >> 32), m, 32);
  return (((unsigned long long)(unsigned)hi) << 32) | (unsigned long long)(unsigned)lo;
}

__global__ __launch_bounds__(MNODES, 1)
void som_train_kernel(const float* __restrict__ data,
                      const float* __restrict__ nodes_in,
                      const int*   __restrict__ rand_idx,
                      float*       __restrict__ out)
{
  __shared__ float xbuf[NBUF][D];            // async-load landing ring (4 x 128B)
  __shared__ unsigned long long warr[32];    // per-wave argmin keys
  __shared__ int idxs[NITER];                // staged rand_indices

  const int tid  = threadIdx.x;              // == node id, 0..1023
  const int lane = tid & 31;

  // ---- prologue: stage indices to LDS, pull my node row into registers ----
  for (int i = tid; i < NITER; i += MNODES) idxs[i] = rand_idx[i];

  float node[D];
  {
    const float4* p = (const float4*)(nodes_in + tid * D);
#pragma unroll
    for (int q = 0; q < D / 4; ++q) {
      float4 v = p[q];
      node[4*q+0] = v.x; node[4*q+1] = v.y; node[4*q+2] = v.z; node[4*q+3] = v.w;
    }
  }
  __syncthreads();

  // ---- issue first PF async prefetches (wave 0, lanes 0..7: 8 x b128 = 128B) ----
  if (tid < 8) {
#pragma unroll
    for (int j = 0; j < PF; ++j) {
      unsigned long long g =
          (unsigned long long)(uintptr_t)(data + (size_t)idxs[j] * D + tid * 4);
      unsigned int l = (unsigned int)(uintptr_t)(&xbuf[j][tid * 4]); // LDS byte offset
      asm volatile("global_load_async_to_lds_b128 %0, %1, off"
                   :: "v"(l), "v"(g) : "memory");
    }
  }

  const int row = tid >> 5;
  const int col = tid & 31;
  float thr = 20.0f;                          // THRESH0, sequential f32 accumulation

  for (int k = 0; k < NITER; ++k) {
    // -- prefetch x[k+PF], then retire x[k]; in-order ASYNC completion means
    //    cnt<=3 after issuing x[k+3] guarantees x[k] has landed in LDS.
    if (k + PF < NITER) {
      if (tid < 8) {
        unsigned long long g =
            (unsigned long long)(uintptr_t)(data + (size_t)idxs[k + PF] * D + tid * 4);
        unsigned int l =
            (unsigned int)(uintptr_t)(&xbuf[(k + PF) & (NBUF - 1)][tid * 4]);
        asm volatile("global_load_async_to_lds_b128 %0, %1, off"
                     :: "v"(l), "v"(g) : "memory");
      }
      asm volatile("s_wait_asynccnt 0x3" ::: "memory");
    } else {
      asm volatile("s_wait_asynccnt 0x0" ::: "memory");   // drain tail
    }
    __syncthreads();   // barrier A: x[k] visible; warr free for rewrite

    // -- broadcast-read x[k] into registers (same address per lane -> LDS bcast)
    float x[D];
    {
      const float4* p = (const float4*)(&xbuf[k & (NBUF - 1)][0]);
#pragma unroll
      for (int q = 0; q < D / 4; ++q) {
        float4 v = p[q];
        x[4*q+0] = v.x; x[4*q+1] = v.y; x[4*q+2] = v.z; x[4*q+3] = v.w;
      }
    }

    // -- squared distance + sqrt (matches reference eucl_dist before argmin)
    float s = 0.0f;
#pragma unroll
    for (int d = 0; d < D; ++d) { float df = x[d] - node[d]; s += df * df; }
    float dist = sqrtf(s);

    // -- argmin with first-min tie-break: u64 key = (dist_bits << 32) | node_id
    unsigned long long key =
        ((unsigned long long)__float_as_uint(dist) << 32) | (unsigned)tid;
#pragma unroll
    for (int off = 16; off; off >>= 1) {
      unsigned long long o = shfl_xor_u64(key, off);
      if (o < key) key = o;
    }
    if (lane == 0) warr[tid >> 5] = key;
    __syncthreads();   // barrier B: warr published

    unsigned long long key2 = warr[lane];   // 32 wave results, one per lane
#pragma unroll
    for (int off = 16; off; off >>= 1) {
      unsigned long long o = shfl_xor_u64(key2, off);
      if (o < key2) key2 = o;
    }
    const unsigned nearest = (unsigned)key2;  // all lanes hold global argmin

    // -- neighborhood update (Chebyshev grid distance computed analytically;
    //    identical to the nhbrdist input: xy[m] = (m/32, m%32))
    const float thr_used = fmaxf(thr, 0.5f);
    const float alpha = 0.05f - 0.04f * ((float)k / (float)NITER);

    int dr = row - (int)(nearest >> 5); dr = dr < 0 ? -dr : dr;
    int dc = col - (int)(nearest & 31); dc = dc < 0 ? -dc : dc;
    const int cheb = dr > dc ? dr : dc;
    if ((float)cheb <= thr_used) {
#pragma unroll
      for (int d = 0; d < D; ++d) node[d] = node[d] + (x[d] - node[d]) * alpha;
    }
    thr = thr_used + (-0.00975f);   // THRESH_STEP, sequential like the scan
  }

  // ---- write final nodes ----
  {
    float4* p = (float4*)(out + tid * D);
#pragma unroll
    for (int q = 0; q < D / 4; ++q) {
      float4 v;
      v.x = node[4*q+0]; v.y = node[4*q+1]; v.z = node[4*q+2]; v.w = node[4*q+3];
      p[q] = v;
    }
  }
}

extern "C" void kernel_launch(void* const* d_in, const int* in_sizes, int n_in,
                              void* d_out, int out_size, void* d_ws, size_t ws_size,
                              hipStream_t stream) {
  const float* data  = (const float*)d_in[0];   // [500000 * 32] f32
  const float* nodes = (const float*)d_in[1];   // [1024 * 32]   f32
  // d_in[2] = nhbrdist [1024*1024] f32 — unused (computed analytically)
  const int*   ridx  = (const int*)d_in[3];     // [2000] i32

  som_train_kernel<<<1, MNODES, 0, stream>>>(data, nodes, ridx, (float*)d_out);
}